// CrossEntropyLoss_58136677319100
// MI455X (gfx1250) — compile-verified
//
#include <hip/hip_runtime.h>
#include <stdint.h>

// Cross-entropy (log-softmax NLL) for N=4096 rows, C=50257 fp32 classes.
// Memory-bound: ~823 MB streamed once -> ~35us floor at 23.3 TB/s.
// One wave32 per row, online logsumexp, CDNA5 async global->LDS streaming
// (GLOBAL_LOAD_ASYNC_TO_LDS_B128 + ASYNCcnt), depth-3 pipeline of 2KB groups
// over 4 LDS buffers (6 KB in flight per wave -> ~24 MB machine-wide, above
// the ~19 MB latency-bandwidth product needed to saturate HBM).
//
// Group issue trick: the instruction's INST_OFFSET is added to BOTH the LDS
// destination and the global source address (CDNA5 ISA, async ch. 4.4), so a
// 2KB group is 4 issues off one (lds,addr) operand pair with offset:0..1536.

#define WAVES_PER_BLOCK 8
#define CHUNK_FLOATS    128          // 32 lanes * 4 floats (b128 per lane)
#define GROUP_CHUNKS    4
#define GROUP_FLOATS    (GROUP_CHUNKS * CHUNK_FLOATS)   // 512 floats = 2 KB
#define GROUP_BYTES     (GROUP_FLOATS * 4)
#define NBUF            4            // buffer ring (power of two for g&3)
#define DEPTH_GROUPS    3            // groups in flight

// Online logsumexp update; exactly ONE v_exp_f32 per element.
// State: m = running max, s = sum of exp(x - m).
__device__ __forceinline__ void ce_update(float& m, float& s, float x) {
    float d  = x - m;                    // +inf on first element (m = -inf)
    float e  = __expf(-fabsf(d));        // exp(-|d|); exp(-inf) -> 0
    bool  gt = d > 0.0f;
    s = gt ? __builtin_fmaf(s, e, 1.0f)  // rescale old sum, add exp(0)
           : (s + e);
    m = fmaxf(m, x);
}

__global__ __launch_bounds__(256)
void ce_loss_kernel(const float* __restrict__ logits,
                    const int*   __restrict__ target,
                    float* __restrict__ out,
                    int N, int C)
{
    // 8 waves * 4 buffers * 2 KB = 64 KB static LDS (WGP has 320 KB)
    __shared__ float smem[WAVES_PER_BLOCK * NBUF * GROUP_FLOATS];

    const int lane = threadIdx.x & 31;
    const int wave = threadIdx.x >> 5;
    const int row  = blockIdx.x * WAVES_PER_BLOCK + wave;
    if (row >= N) return;                         // uniform per wave

    const float* rowp = logits + (size_t)row * (size_t)C;

    float m = -__builtin_inff();
    float s = 0.0f;

    // ---- alignment head peel: rows are only 4B aligned (C*4 % 16 != 0) ----
    uintptr_t base = (uintptr_t)rowp;
    int head = (int)(((16u - (unsigned)(base & 15u)) & 15u) >> 2); // 0..3 floats
    if (head > C) head = C;

    const int main_n  = C - head;
    const int ngroups = main_n / GROUP_FLOATS;           // 98 groups of 2 KB
    const bool piped  = (ngroups >= DEPTH_GROUPS);
    const int tstart  = piped ? (head + ngroups * GROUP_FLOATS) : head;

    // Per-lane LDS destination (wave-relative byte offset; low 32 bits of the
    // generic shared pointer are the DS-relative LDS address).
    const unsigned ldsLane =
        (unsigned)(uintptr_t)(&smem[wave * NBUF * GROUP_FLOATS]) +
        (unsigned)lane * 16u;
    // Per-lane global source for the aligned main region.
    const unsigned long long gLane =
        (unsigned long long)(base + (uintptr_t)head * 4u + (uintptr_t)lane * 16u);

    // Issue one 2 KB group (4 async b128 per lane-slot) into buffer (g & 3).
    auto issue_group = [&](int g) {
        unsigned           l  = ldsLane + (unsigned)(g & (NBUF - 1)) * GROUP_BYTES;
        unsigned long long gp = gLane + (unsigned long long)g * GROUP_BYTES;
        asm volatile(
            "global_load_async_to_lds_b128 %0, %1, off\n\t"
            "global_load_async_to_lds_b128 %0, %1, off offset:512\n\t"
            "global_load_async_to_lds_b128 %0, %1, off offset:1024\n\t"
            "global_load_async_to_lds_b128 %0, %1, off offset:1536"
            :: "v"(l), "v"(gp) : "memory");
    };

    // Consume one group from LDS: 4 x ds_load_b128, 16 online updates.
    auto process_group = [&](int g) {
        const float* lg =
            &smem[(wave * NBUF + (g & (NBUF - 1))) * GROUP_FLOATS + lane * 4];
        #pragma unroll
        for (int c = 0; c < GROUP_CHUNKS; ++c) {
            const float4 v = *(const float4*)(lg + c * CHUNK_FLOATS);
            ce_update(m, s, v.x);
            ce_update(m, s, v.y);
            ce_update(m, s, v.z);
            ce_update(m, s, v.w);
        }
    };

    if (piped) {
        issue_group(0);
        issue_group(1);
        issue_group(2);                       // 12 async ops in flight
    }
    // Overlap: process the unaligned head while the first loads fly.
    if (lane < head) ce_update(m, s, rowp[lane]);

    if (piped) {
        for (int g = 0; g + DEPTH_GROUPS < ngroups; ++g) {
            // in-order completion: <=8 outstanding => group g fully in LDS
            asm volatile("s_wait_asynccnt 0x8" ::: "memory");
            process_group(g);
            // ds reads must retire before the async engine re-arms a buffer
            // (async LDS writes are unordered vs this wave's DS accesses)
            asm volatile("s_wait_dscnt 0x0" ::: "memory");
            issue_group(g + DEPTH_GROUPS);
        }
        asm volatile("s_wait_asynccnt 0x8" ::: "memory");
        process_group(ngroups - 3);
        asm volatile("s_wait_asynccnt 0x4" ::: "memory");
        process_group(ngroups - 2);
        asm volatile("s_wait_asynccnt 0x0" ::: "memory");
        process_group(ngroups - 1);
    }

    // ---- scalar tail (or whole row if too short to pipeline) ----
    for (int i = tstart + lane; i < C; i += 32)
        ce_update(m, s, rowp[i]);

    // ---- wave32 reduction of (m, s) pairs ----
    #pragma unroll
    for (int off = 16; off > 0; off >>= 1) {
        float m2 = __shfl_xor(m, off, 32);
        float s2 = __shfl_xor(s, off, 32);
        float mn = fmaxf(m, m2);
        s = s * __expf(m - mn) + s2 * __expf(m2 - mn);
        m = mn;
    }

    if (lane == 0) {
        int   t  = target[row];
        float xt = rowp[t];
        out[row] = (m + __logf(s)) - xt;   // -(x_t - m - log(sum))
    }
}

extern "C" void kernel_launch(void* const* d_in, const int* in_sizes, int n_in,
                              void* d_out, int out_size, void* d_ws, size_t ws_size,
                              hipStream_t stream) {
    const float* logits = (const float*)d_in[0];
    const int*   target = (const int*)d_in[1];
    float*       out    = (float*)d_out;

    const int N = in_sizes[1];                 // 4096 rows (target count)
    const int C = in_sizes[0] / N;             // 50257 classes

    dim3 grid((N + WAVES_PER_BLOCK - 1) / WAVES_PER_BLOCK); // 512 blocks
    dim3 block(WAVES_PER_BLOCK * 32);                       // 256 threads
    ce_loss_kernel<<<grid, block, 0, stream>>>(logits, target, out, N, C);
}